// HGTLayer_2336462209111
// MI455X (gfx1250) — compile-verified
//
#include <hip/hip_runtime.h>
#include <hip/hip_bf16.h>

// Problem constants (fixed by the reference).
#define NN 20000
#define EE 160000
#define DD 128
#define HH 8
#define NF ((size_t)NN * DD)        // 2,560,000 floats per node-feature array
#define N8 ((size_t)NN * HH)        // 160,000  per-node-head scalars
#define EH8 ((size_t)EE * HH)       // 1,280,000 per-edge-head scalars

typedef float v2f __attribute__((ext_vector_type(2)));
typedef float v8f __attribute__((ext_vector_type(8)));
typedef unsigned int u32x4 __attribute__((ext_vector_type(4)));
typedef int i32x4 __attribute__((ext_vector_type(4)));
typedef int i32x8 __attribute__((ext_vector_type(8)));

// ---------------- helpers ----------------

// Order-preserving float<->uint encoding so atomicMax(u32) works for floats.
__device__ __forceinline__ unsigned encf(float f) {
    unsigned u = __float_as_uint(f);
    return (u >> 31) ? ~u : (u | 0x80000000u);
}
__device__ __forceinline__ float decf(unsigned u) {
    return __uint_as_float((u >> 31) ? (u & 0x7FFFFFFFu) : ~u);
}
__device__ __forceinline__ float guard(float s) { return (s == 0.0f) ? 1.0f : s; }

__device__ __forceinline__ float dot16(const float* __restrict__ a,
                                       const float* __restrict__ b) {
    const float4* A = (const float4*)a;
    const float4* B = (const float4*)b;
    float s = 0.0f;
#pragma unroll
    for (int i = 0; i < 4; ++i) {
        float4 x = A[i], y = B[i];
        s += x.x * y.x + x.y * y.y + x.z * y.z + x.w * y.w;
    }
    return s;
}

// ---------------- kernel 0: init scratch ----------------
__global__ void init_ws(float* agg, float* nodesum, unsigned* nodemax) {
    size_t i = (size_t)blockIdx.x * blockDim.x + threadIdx.x;
    if (i < 3 * NF) agg[i] = 0.0f;
    if (i < 5 * N8) { nodesum[i] = 0.0f; nodemax[i] = 0u; }
}

// ---------------- kernel 1: fold per-head 16x16 relation mats into weights ----
// out[j, h*16+f] = sum_d W[j, h*16+d] * M[h,d,f]; same fold for the bias row.
__global__ void combine_weights(const float* __restrict__ Wk, const float* __restrict__ bk,
                                const float* __restrict__ Wv, const float* __restrict__ bv,
                                const float* __restrict__ rel_att,
                                const float* __restrict__ wm0, const float* __restrict__ wm1,
                                const float* __restrict__ wm2,
                                const float* __restrict__ rel_msg,
                                float* Wka, float* bka, float* Wkw, float* bkw,
                                float* Wvm, float* bvm) {
    int which = blockIdx.y;          // 0: rel_att, 1..3: wm[m], 4: rel_msg
    int row   = blockIdx.x;          // 0..127 matrix rows, 128 == bias row
    int col   = threadIdx.x;         // 0..127
    const float *M, *Ws, *bs; float *Wo, *bo;
    switch (which) {
        case 0:  M = rel_att; Ws = Wk; bs = bk; Wo = Wka;             bo = bka;       break;
        case 1:  M = wm0;     Ws = Wk; bs = bk; Wo = Wkw;             bo = bkw;       break;
        case 2:  M = wm1;     Ws = Wk; bs = bk; Wo = Wkw + 16384;     bo = bkw + 128; break;
        case 3:  M = wm2;     Ws = Wk; bs = bk; Wo = Wkw + 2 * 16384; bo = bkw + 256; break;
        default: M = rel_msg; Ws = Wv; bs = bv; Wo = Wvm;             bo = bvm;       break;
    }
    int h = col >> 4, f = col & 15;
    float s = 0.0f;
    if (row < 128) {
#pragma unroll
        for (int d = 0; d < 16; ++d)
            s += Ws[row * 128 + h * 16 + d] * M[h * 256 + d * 16 + f];
        Wo[row * 128 + col] = s;
    } else {
#pragma unroll
        for (int d = 0; d < 16; ++d)
            s += bs[h * 16 + d] * M[h * 256 + d * 16 + f];
        bo[col] = s;
    }
}

// ---------------- kernel 2: WMMA fp32 GEMM  C[n,128] = X[n,128]@W[128,128]+b --
// The 16x128 A-strip is staged into LDS once per block via the Tensor Data
// Mover (6-arg builtin on this toolchain), then shared by all 8 waves.
// Wave w owns 16-column tile w. Optional epilogue:
// C = C*sigmoid(skip) + resid*(1-sigmoid(skip)).
__global__ void gemm128_wmma(const float* __restrict__ X, const float* __restrict__ W,
                             const float* __restrict__ bias, float* __restrict__ C,
                             const float* __restrict__ resid,
                             const float* __restrict__ skipPtr) {
    __shared__ float As[16 * DD];                  // 8 KB A-strip
    const int row0 = blockIdx.x * 16;
    const int wave = threadIdx.x >> 5;
    const int lane = threadIdx.x & 31;
    const int l15  = lane & 15;
    const int hi   = lane >> 4;          // 0: K 0/1, 1: K 2/3 (ISA 16x4 layout)
    const int col0 = wave * 16;

#if __has_builtin(__builtin_amdgcn_tensor_load_to_lds) && \
    __has_builtin(__builtin_amdgcn_s_wait_tensorcnt)
    if (threadIdx.x < 32) {
        // Tensor DMA descriptor (ISA 8.3/8.4): 2-D tensor, 4-byte elements,
        // tile = 128 x 16 rows starting at X[row0][0], packed into LDS.
        unsigned long long ga = (unsigned long long)(const void*)(X + (size_t)row0 * DD);
        unsigned lds_off = (unsigned)(uintptr_t)(void*)As;  // LDS aperture: addr[31:0]
        u32x4 g0;
        g0.x = 0x1u;                                        // count=1, no gather
        g0.y = lds_off;                                     // lds_addr
        g0.z = (unsigned)(ga & 0xFFFFFFFFull);              // global_addr[31:0]
        g0.w = (unsigned)((ga >> 32) & 0x01FFFFFFull)       // global_addr[56:32]
               | (2u << 30);                                // type = 2 ("image")
        i32x8 g1;
        g1[0] = 0x00020000;                                 // data_size = 4B
        g1[1] = (int)(128u << 16);                          // tensor_dim0 = 128
        g1[2] = (int)((unsigned)NN << 16);                  // tensor_dim1 = 20000
        g1[3] = (int)(128u << 16);                          // tile_dim0 = 128
        g1[4] = 16;                                         // tile_dim1 = 16
        g1[5] = 128;                                        // tensor_dim0_stride = 128
        g1[6] = 0;
        g1[7] = 0;
        i32x4 z4 = {0, 0, 0, 0};
        i32x8 z8 = {0, 0, 0, 0, 0, 0, 0, 0};
        __builtin_amdgcn_tensor_load_to_lds(g0, g1, z4, z4, z8, 0);
        __builtin_amdgcn_s_wait_tensorcnt(0);
    }
#else
    // Fallback: cooperative load of the contiguous 16x128 strip.
    {
        const float4* srcv = (const float4*)(X + (size_t)row0 * DD);
        float4* dstv = (float4*)As;
        for (int i = threadIdx.x; i < 16 * DD / 4; i += 256) dstv[i] = srcv[i];
    }
#endif
    __syncthreads();

    v8f acc = {0.f, 0.f, 0.f, 0.f, 0.f, 0.f, 0.f, 0.f};
    const float* arow = As + l15 * DD;             // LDS row for this lane's M
#pragma unroll 8
    for (int kk = 0; kk < DD; kk += 4) {
        v2f a, b;
        a.x = arow[kk + 2 * hi];
        a.y = arow[kk + 2 * hi + 1];
        b.x = W[(size_t)(kk + 2 * hi) * DD + col0 + l15];
        b.y = W[(size_t)(kk + 2 * hi + 1) * DD + col0 + l15];
        acc = __builtin_amdgcn_wmma_f32_16x16x4_f32(false, a, false, b,
                                                    (short)0, acc, false, false);
    }

    const float bvv = bias[col0 + l15];
    float alpha = 1.0f, oneMinus = 0.0f;
    if (resid) {
        float sk = skipPtr[0];
        alpha = 1.0f / (1.0f + __expf(-sk));
        oneMinus = 1.0f - alpha;
    }
#pragma unroll
    for (int r = 0; r < 8; ++r) {
        int row = row0 + r + 8 * hi;               // D VGPR r: rows r / r+8
        size_t o = (size_t)row * DD + col0 + l15;
        float val = acc[r] + bvv;
        if (resid) val = val * alpha + resid[o] * oneMinus;
        C[o] = val;
    }
}

// ---------------- kernel 3: per-edge scores + modality softmax + align -------
__global__ void edge_scores(const float* __restrict__ Ka, const float* __restrict__ Kw,
                            const float* __restrict__ Q,
                            const int* __restrict__ src, const int* __restrict__ dst,
                            const float* __restrict__ rel_pri,
                            float* attB, float* lamB, float* alignB,
                            unsigned* nodemax) {
    size_t idx = (size_t)blockIdx.x * blockDim.x + threadIdx.x;
    if (idx >= EH8) return;
    int e = (int)(idx >> 3), h = (int)(idx & 7);
    int s = src[e], d = dst[e];
    size_t so  = (size_t)s * DD + h * 16;
    size_t dof = (size_t)d * DD + h * 16;
    float pri = rel_pri[h] * 2.5f;    // rel_pri / sqrt(16) * 10

    float att[3], lam[3];
#pragma unroll
    for (int m = 0; m < 3; ++m) {
        const float* q = Q + m * NF + dof;
        att[m] = dot16(Ka + m * NF + so, q) * pri;
        lam[m] = dot16(Kw + m * NF + so, q);
    }
    // softmax over the 3 modalities
    float mx = fmaxf(lam[0], fmaxf(lam[1], lam[2]));
    float e0 = __expf(lam[0] - mx), e1 = __expf(lam[1] - mx), e2 = __expf(lam[2] - mx);
    float inv = 1.0f / (e0 + e1 + e2);
    size_t eo = (size_t)e * HH + h;
    lamB[eo] = e0 * inv; lamB[EH8 + eo] = e1 * inv; lamB[2 * EH8 + eo] = e2 * inv;

    float align = -(fabsf(att[0] - att[1]) + fabsf(att[0] - att[2]) +
                    fabsf(att[1] - att[2])) * (1.0f / 3.0f);
    size_t no = (size_t)d * HH + h;
#pragma unroll
    for (int m = 0; m < 3; ++m) {
        attB[m * EH8 + eo] = att[m];
        atomicMax(&nodemax[m * N8 + no], encf(att[m]));
    }
    alignB[eo] = align;
    atomicMax(&nodemax[3 * N8 + no], encf(align));
}

// ---------------- kernel 4: exp(x - segmax) and segment sums -----------------
__global__ void edge_exp(const int* __restrict__ dst, float* attB, float* alignB,
                         const unsigned* __restrict__ nodemax, float* nodesum) {
    size_t idx = (size_t)blockIdx.x * blockDim.x + threadIdx.x;
    if (idx >= EH8) return;
    int e = (int)(idx >> 3), h = (int)(idx & 7);
    size_t eo = (size_t)e * HH + h;
    size_t no = (size_t)dst[e] * HH + h;
#pragma unroll
    for (int m = 0; m < 3; ++m) {
        float ev = __expf(attB[m * EH8 + eo] - decf(nodemax[m * N8 + no]));
        attB[m * EH8 + eo] = ev;
        atomicAdd(&nodesum[m * N8 + no], ev);
    }
    float ev = __expf(alignB[eo] - decf(nodemax[3 * N8 + no]));
    alignB[eo] = ev;
    atomicAdd(&nodesum[3 * N8 + no], ev);
}

// ---------------- kernel 5: beta1*beta2 and its segment max ------------------
__global__ void edge_beta1(const int* __restrict__ dst,
                           const float* __restrict__ attB, const float* __restrict__ lamB,
                           const float* __restrict__ alignB,
                           const float* __restrict__ nodesum, unsigned* nodemax,
                           float* betaB) {
    size_t idx = (size_t)blockIdx.x * blockDim.x + threadIdx.x;
    if (idx >= EH8) return;
    int e = (int)(idx >> 3), h = (int)(idx & 7);
    size_t eo = (size_t)e * HH + h;
    size_t no = (size_t)dst[e] * HH + h;
    float beta2 = alignB[eo] / guard(nodesum[3 * N8 + no]);
    float b1 = 0.0f;
#pragma unroll
    for (int m = 0; m < 3; ++m)
        b1 += (attB[m * EH8 + eo] / guard(nodesum[m * N8 + no])) * lamB[m * EH8 + eo];
    float sc = b1 * beta2;
    betaB[eo] = sc;
    atomicMax(&nodemax[4 * N8 + no], encf(sc));
}

// ---------------- kernel 6: final softmax exp + sum --------------------------
__global__ void edge_exp2(const int* __restrict__ dst, float* betaB,
                          const unsigned* __restrict__ nodemax, float* nodesum) {
    size_t idx = (size_t)blockIdx.x * blockDim.x + threadIdx.x;
    if (idx >= EH8) return;
    int e = (int)(idx >> 3), h = (int)(idx & 7);
    size_t eo = (size_t)e * HH + h;
    size_t no = (size_t)dst[e] * HH + h;
    float ev = __expf(betaB[eo] - decf(nodemax[4 * N8 + no]));
    betaB[eo] = ev;
    atomicAdd(&nodesum[4 * N8 + no], ev);
}

// ---------------- kernel 7: beta-weighted scatter-add of messages ------------
__global__ void edge_aggregate(const int* __restrict__ src, const int* __restrict__ dst,
                               const float* __restrict__ betaB,
                               const float* __restrict__ nodesum,
                               const float* __restrict__ Vr, float* agg) {
    size_t idx = (size_t)blockIdx.x * blockDim.x + threadIdx.x;
    if (idx >= EH8) return;
    int e = (int)(idx >> 3), h = (int)(idx & 7);
    size_t eo = (size_t)e * HH + h;
    size_t no = (size_t)dst[e] * HH + h;
    float beta = betaB[eo] / guard(nodesum[4 * N8 + no]);
    size_t so  = (size_t)src[e] * DD + h * 16;
    size_t dof = (size_t)dst[e] * DD + h * 16;
#pragma unroll
    for (int m = 0; m < 3; ++m) {
        const float* v = Vr + m * NF + so;
        float* a = agg + m * NF + dof;
#pragma unroll
        for (int f = 0; f < 16; ++f)
            atomicAdd(&a[f], beta * v[f]);
    }
}

// ---------------- host-side launcher ----------------
extern "C" void kernel_launch(void* const* d_in, const int* in_sizes, int n_in,
                              void* d_out, int out_size, void* d_ws, size_t ws_size,
                              hipStream_t stream) {
    const float* x[3] = {(const float*)d_in[0], (const float*)d_in[1], (const float*)d_in[2]};
    const float* Wk = (const float*)d_in[3];
    const float* bk = (const float*)d_in[4];
    const float* Wq = (const float*)d_in[5];
    const float* bq = (const float*)d_in[6];
    const float* Wv = (const float*)d_in[7];
    const float* bv = (const float*)d_in[8];
    const float* Wa = (const float*)d_in[9];
    const float* ba = (const float*)d_in[10];
    const float* rel_att = (const float*)d_in[11];
    const float* rel_msg = (const float*)d_in[12];
    const float* rel_pri = (const float*)d_in[13];
    const float* wm0 = (const float*)d_in[14];
    const float* wm1 = (const float*)d_in[15];
    const float* wm2 = (const float*)d_in[16];
    const float* skip = (const float*)d_in[17];
    const int* src = (const int*)d_in[18];
    const int* dst = (const int*)d_in[19];
    float* out = (float*)d_out;

    // workspace carve-up (floats)
    float* ws     = (float*)d_ws;
    float* Q      = ws;                      // 3*NF
    float* Ka     = Q + 3 * NF;              // 3*NF
    float* Kw     = Ka + 3 * NF;             // 3*NF
    float* Vr     = Kw + 3 * NF;             // 3*NF
    float* agg    = Vr + 3 * NF;             // 3*NF
    float* Wka    = agg + 3 * NF;            // 16384
    float* Wkw    = Wka + 16384;             // 3*16384
    float* Wvm    = Wkw + 3 * 16384;         // 16384
    float* bka    = Wvm + 16384;             // 128
    float* bkw    = bka + 128;               // 384
    float* bvm    = bkw + 384;               // 128
    float* attB   = bvm + 128;               // 3*EH8
    float* lamB   = attB + 3 * EH8;          // 3*EH8
    float* alignB = lamB + 3 * EH8;          // EH8
    float* betaB  = alignB + EH8;            // EH8
    float* nodesum = betaB + EH8;            // 5*N8
    unsigned* nodemax = (unsigned*)(nodesum + 5 * N8); // 5*N8

    // 0) init accumulators / segment max / segment sum
    {
        size_t total = 3 * NF;
        int blocks = (int)((total + 255) / 256);
        init_ws<<<blocks, 256, 0, stream>>>(agg, nodesum, nodemax);
    }
    // 1) fold relation matrices into weights
    combine_weights<<<dim3(129, 5), 128, 0, stream>>>(
        Wk, bk, Wv, bv, rel_att, wm0, wm1, wm2, rel_msg,
        Wka, bka, Wkw, bkw, Wvm, bvm);

    // 2) twelve projection GEMMs (WMMA + TDM-staged A operand)
    const int gblocks = NN / 16;  // 1250, exact
    for (int m = 0; m < 3; ++m) {
        gemm128_wmma<<<gblocks, 256, 0, stream>>>(x[m], Wq, bq, Q + m * NF, nullptr, nullptr);
        gemm128_wmma<<<gblocks, 256, 0, stream>>>(x[m], Wka, bka, Ka + m * NF, nullptr, nullptr);
        gemm128_wmma<<<gblocks, 256, 0, stream>>>(x[m], Wkw + m * 16384, bkw + m * 128,
                                                  Kw + m * NF, nullptr, nullptr);
        gemm128_wmma<<<gblocks, 256, 0, stream>>>(x[m], Wvm, bvm, Vr + m * NF, nullptr, nullptr);
    }

    // 3..7) edge pipeline
    const int eblocks = (int)((EH8 + 255) / 256);
    edge_scores<<<eblocks, 256, 0, stream>>>(Ka, Kw, Q, src, dst, rel_pri,
                                             attB, lamB, alignB, nodemax);
    edge_exp<<<eblocks, 256, 0, stream>>>(dst, attB, alignB, nodemax, nodesum);
    edge_beta1<<<eblocks, 256, 0, stream>>>(dst, attB, lamB, alignB, nodesum, nodemax, betaB);
    edge_exp2<<<eblocks, 256, 0, stream>>>(dst, betaB, nodemax, nodesum);
    edge_aggregate<<<eblocks, 256, 0, stream>>>(src, dst, betaB, nodesum, Vr, agg);

    // 8) output GEMMs with sigmoid-skip residual epilogue
    for (int m = 0; m < 3; ++m)
        gemm128_wmma<<<gblocks, 256, 0, stream>>>(agg + m * NF, Wa, ba,
                                                  out + m * NF, x[m], skip);
}